// SampledSoftmaxLoss_45621142618832
// MI455X (gfx1250) — compile-verified
//
#include <hip/hip_runtime.h>

typedef __attribute__((ext_vector_type(2))) float v2f;
typedef __attribute__((ext_vector_type(8))) float v8f;

#define DIM        128
#define NUM_NEG    64
#define NUM_CAND   65     // 1 positive + 64 negatives
#define TILE       16     // batch rows per workgroup / WMMA group
#define WAVES      8      // waves per workgroup (256 threads, wave32)

// Kernel 1: per 16-row batch tile, compute all 65 logits per row via
// diagonal-extraction WMMA (fp32 16x16x4), then per-row logsumexp - pos,
// and one partial sum per workgroup into d_ws.
__global__ __launch_bounds__(TILE * 2 * WAVES) void nce_logits_kernel(
    const float* __restrict__ x,          // [B, DIM]
    const float* __restrict__ W,          // [V, DIM]
    const float* __restrict__ log_q,      // [V]
    const int*   __restrict__ true_labels,// [B]
    const int*   __restrict__ neg_indices,// [B, NUM_NEG]
    float*       __restrict__ partial)    // [gridDim.x]
{
    __shared__ float s_scores[TILE][NUM_CAND + 1];        // logits per row (padded)
    __shared__ float s_cmat[WAVES][TILE][TILE + 1];       // per-wave C-matrix staging
    __shared__ float s_row[TILE];                         // per-row loss terms

    const int tid  = threadIdx.x;
    const int wave = tid >> 5;
    const int lane = tid & 31;
    const int p    = lane & 15;     // pair index within group (== M == N on diag)
    const int half = lane >> 4;     // lane half selects K sub-block
    const int b    = blockIdx.x * TILE + p;

    // Per the 16x16x4 f32 WMMA layout, lane L supplies
    //   A[M=p][kb + 2*half + {0,1}]  and  B[kb + 2*half + {0,1}][N=p]
    // -> identical float2 addressing into the x row and the gathered W row.
    const float* xrow = x + (size_t)b * DIM + 2 * half;

    // Candidate loop: c depends only on wave id + iteration -> wave-uniform,
    // EXEC is all-1s at every WMMA.
    for (int c = wave; c < NUM_CAND; c += WAVES) {
        const int idx = (c == 0) ? true_labels[b]
                                 : neg_indices[(size_t)b * NUM_NEG + (c - 1)];
        const float* wrow = W + (size_t)idx * DIM + 2 * half;

        v8f acc = {};
#pragma unroll
        for (int t = 0; t < DIM / 4; ++t) {
            v2f a  = *(const v2f*)(xrow + 4 * t);
            v2f bv = *(const v2f*)(wrow + 4 * t);
            // 8 args: (neg_a, A, neg_b, B, c_mod, C, reuse_a, reuse_b)
            acc = __builtin_amdgcn_wmma_f32_16x16x4_f32(
                false, a, false, bv, (short)0, acc, false, false);
        }

        // C layout: lane L, vgpr v holds C[M][N], N = L%16, M = v + 8*(L/16).
        // Stage to LDS and read back the diagonal (DS ops in a wave are in-order).
#pragma unroll
        for (int v = 0; v < 8; ++v) {
            s_cmat[wave][v + 8 * half][p] = acc[v];
        }
        if (lane < TILE) {
            s_scores[lane][c] = s_cmat[wave][lane][lane] - log_q[idx];
        }
    }

    __syncthreads();

    // Per-row logsumexp over 65 logits; target class is column 0.
    if (tid < TILE) {
        float mx = s_scores[tid][0];
#pragma unroll 4
        for (int c = 1; c < NUM_CAND; ++c) {
            float v = s_scores[tid][c];
            mx = v > mx ? v : mx;
        }
        float sum = 0.0f;
#pragma unroll 4
        for (int c = 0; c < NUM_CAND; ++c) {
            sum += __expf(s_scores[tid][c] - mx);
        }
        float lse = mx + __logf(sum);
        s_row[tid] = lse - s_scores[tid][0];
    }
    __syncthreads();

    if (tid == 0) {
        float acc = 0.0f;
#pragma unroll
        for (int r = 0; r < TILE; ++r) acc += s_row[r];
        partial[blockIdx.x] = acc;
    }
}

// Kernel 2: deterministic single-block reduction of the per-tile partials.
__global__ __launch_bounds__(256) void nce_reduce_kernel(
    const float* __restrict__ partial, float* __restrict__ out,
    int n, float inv_total)
{
    __shared__ float s[256];
    const int t = threadIdx.x;
    float v = 0.0f;
    for (int i = t; i < n; i += 256) v += partial[i];   // fixed order per thread
    s[t] = v;
    __syncthreads();
    for (int st = 128; st > 0; st >>= 1) {
        if (t < st) s[t] += s[t + st];
        __syncthreads();
    }
    if (t == 0) out[0] = s[0] * inv_total;
}

extern "C" void kernel_launch(void* const* d_in, const int* in_sizes, int n_in,
                              void* d_out, int out_size, void* d_ws, size_t ws_size,
                              hipStream_t stream) {
    (void)n_in; (void)out_size; (void)ws_size;

    const float* x           = (const float*)d_in[0];  // [B, 128]
    const float* W           = (const float*)d_in[1];  // [V, 128]
    const float* log_q       = (const float*)d_in[2];  // [V]
    const int*   true_labels = (const int*)d_in[3];    // [B]
    const int*   neg_indices = (const int*)d_in[4];    // [B, 64]

    const int B      = in_sizes[0] / DIM;              // 8192
    const int ntiles = B / TILE;                       // 512

    float* partial = (float*)d_ws;                     // ntiles floats of scratch

    nce_logits_kernel<<<ntiles, TILE * 2 * WAVES, 0, stream>>>(
        x, W, log_q, true_labels, neg_indices, partial);

    nce_reduce_kernel<<<1, 256, 0, stream>>>(
        partial, (float*)d_out, ntiles, 1.0f / (float)B);
}